// AdaSpatialMLP_82102594830443
// MI455X (gfx1250) — compile-verified
//
#include <hip/hip_runtime.h>

typedef float v2f __attribute__((ext_vector_type(2)));
typedef float v8f __attribute__((ext_vector_type(8)));
typedef unsigned int u32x4 __attribute__((ext_vector_type(4)));
typedef int i32x4 __attribute__((ext_vector_type(4)));
typedef int i32x8 __attribute__((ext_vector_type(8)));

// Problem constants (from reference): B=128, N=196, DIM=384, K=16, H=6, R=4
#define BSZ   128
#define NTOK  196
#define CDIM  384
#define KB    16
#define HH    6
#define KH    96   // K*H
#define HD    64   // DIM/H

// ---- Tensor Data Mover support (6-arg clang-23 builtin; fallback = VALU copy)
#if defined(__has_builtin)
#if __has_builtin(__builtin_amdgcn_tensor_load_to_lds) && \
    __has_builtin(__builtin_amdgcn_s_wait_tensorcnt)
#define HAVE_TDM 1
#endif
#endif

#ifdef HAVE_TDM
// 1D contiguous copy of len_f32 floats (len <= 65535) into LDS byte offset.
__device__ __forceinline__ void tdm_load_1d(unsigned lds_off, const void* gptr,
                                            unsigned len_f32) {
    const unsigned long long ga = (unsigned long long)(uintptr_t)gptr;
    u32x4 g0 = { 1u,                                   // count=1 valid descriptor
                 lds_off,                               // lds_addr (bytes)
                 (unsigned)(ga & 0xFFFFFFFFu),          // global_addr[31:0]
                 (unsigned)((ga >> 32) & 0x01FFFFFFu) | 0x80000000u }; // [56:32]|type=2
    i32x8 g1;
    g1[0] = (2 << 16);                                  // data_size = 4 bytes
    g1[1] = (int)((len_f32 & 0xFFFFu) << 16);           // tensor_dim0[15:0]
    g1[2] = (int)((len_f32 >> 16) | (1u << 16));        // tensor_dim0[31:16]|tensor_dim1=1
    g1[3] = (int)(len_f32 << 16);                       // tile_dim0 = len
    g1[4] = 1;                                          // tile_dim1 = 1
    g1[5] = (int)len_f32;                               // tensor_dim0_stride
    g1[6] = 0; g1[7] = 0;
    const i32x4 z4 = {0, 0, 0, 0};
    const i32x8 z8 = {0, 0, 0, 0, 0, 0, 0, 0};
    __builtin_amdgcn_tensor_load_to_lds(g0, g1, z4, z4, z8, 0);
}

// 2D tile (cols x rows, row stride in floats) with LDS padding:
// pad 'amt' code dwords after every 'ivl' code interval (ivl=5 -> 64 dw, amt=3 -> 4 dw).
__device__ __forceinline__ void tdm_load_2d_pad(unsigned lds_off, const void* gptr,
                                                unsigned cols, unsigned rows,
                                                unsigned stride_f32,
                                                unsigned pad_ivl, unsigned pad_amt) {
    const unsigned long long ga = (unsigned long long)(uintptr_t)gptr;
    u32x4 g0 = { 1u, lds_off,
                 (unsigned)(ga & 0xFFFFFFFFu),
                 (unsigned)((ga >> 32) & 0x01FFFFFFu) | 0x80000000u };
    i32x8 g1;
    g1[0] = (int)((2u << 16) | (1u << 20) | (pad_ivl << 22) | (pad_amt << 25));
    g1[1] = (int)((cols & 0xFFFFu) << 16);              // tensor_dim0 lo
    g1[2] = (int)((cols >> 16) | (rows << 16));         // tensor_dim0 hi | tensor_dim1
    g1[3] = (int)((rows >> 16) | (cols << 16));         // tensor_dim1 hi | tile_dim0
    g1[4] = (int)rows;                                  // tile_dim1 = rows, tile_dim2=0
    g1[5] = (int)stride_f32;                            // tensor_dim0_stride
    g1[6] = 0; g1[7] = 0;
    const i32x4 z4 = {0, 0, 0, 0};
    const i32x8 z8 = {0, 0, 0, 0, 0, 0, 0, 0};
    __builtin_amdgcn_tensor_load_to_lds(g0, g1, z4, z4, z8, 0);
}
#endif

// LDS byte offsets (dynamic LDS assumed at offset 0)
#define K1_XS_OFF   0u
#define K1_W1S_OFF  (64u * 68u * 4u)
#define K1_W2S_OFF  ((64u * 68u + 64u * 96u + 64u * 98u) * 4u)
#define K2_XS_OFF   (38400u * 4u)

// ---------------------------------------------------------------------------
// Kernel 1: mix = softmax_k( gelu(x @ W1 + b1) @ W2 + b2 )   -> ws [B*N, 96]
// 392 blocks x 256 threads (8 waves); 64 rows per block; WMMA f32 16x16x4.
// ---------------------------------------------------------------------------
__global__ __launch_bounds__(256) void mix_mlp_kernel(
    const float* __restrict__ x, const float* __restrict__ W1,
    const float* __restrict__ b1, const float* __restrict__ W2,
    const float* __restrict__ b2, float* __restrict__ mixo)
{
    extern __shared__ float smem[];
    float* xs  = smem;                      // [64][68]  x K-chunk (padded rows)
    float* w1s = xs  + 64 * 68;             // [64][96]  W1 K-chunk; reused: logits/probs
    float* hid = w1s + 64 * 96;             // [64][98]  gelu hidden
    float* w2s = hid + 64 * 98;             // [96][96]  full W2

    const int t    = threadIdx.x;
    const int wave = t >> 5;
    const int lane = t & 31;
    const int half = lane >> 4;             // K0/K1 half vs K2/K3 half
    const int lr   = lane & 15;
    const int row0 = blockIdx.x * 64;

    const int mt  = wave & 3;               // M 16-tile 0..3
    const int ng0 = (wave >> 2) * 3;        // N 16-tiles ng0..ng0+2

#ifdef HAVE_TDM
    if (wave == 0)                          // async DMA: whole W2 -> LDS
        tdm_load_1d(K1_W2S_OFF, W2, 96 * 96);
#else
    #pragma unroll
    for (int i = 0; i < 9; ++i) {
        int f = t + 256 * i;
        ((float4*)w2s)[f] = ((const float4*)W2)[f];
    }
#endif

    // ---------------- GEMM1: hid = x_tile[64,384] @ W1[384,96] --------------
    v8f acc0 = {}, acc1 = {}, acc2 = {};
    for (int c0 = 0; c0 < 6; ++c0) {
        const int kc = c0 * 64;
        __syncthreads();                    // previous chunk fully consumed
#ifdef HAVE_TDM
        if (wave == 0) {
            // x chunk [64 rows][64 cols], row stride 384, DMA-padded to 68-f rows
            tdm_load_2d_pad(K1_XS_OFF, x + (size_t)row0 * CDIM + kc,
                            64, 64, CDIM, 5u /*64 dw*/, 3u /*4 dw*/);
            // W1 chunk: 64 contiguous rows of 96
            tdm_load_1d(K1_W1S_OFF, W1 + (size_t)kc * KH, 64 * KH);
            __builtin_amdgcn_s_wait_tensorcnt(0);
        }
#else
        #pragma unroll
        for (int i = 0; i < 4; ++i) {
            int f = t + 256 * i;
            int r = f >> 4, c4 = f & 15;
            float4 v = ((const float4*)(x + (size_t)(row0 + r) * CDIM + kc))[c4];
            *(float4*)(xs + r * 68 + c4 * 4) = v;
        }
        #pragma unroll
        for (int i = 0; i < 6; ++i) {
            int f = t + 256 * i;
            int r = f / 24, c4 = f % 24;
            ((float4*)(w1s + r * 96))[c4] =
                ((const float4*)(W1 + (size_t)(kc + r) * KH))[c4];
        }
#endif
        __syncthreads();
        #pragma unroll 4
        for (int kk = 0; kk < 64; kk += 4) {
            const int ka = kk + 2 * half;
            v2f a = *(const v2f*)(xs + (mt * 16 + lr) * 68 + ka);
            #pragma unroll
            for (int j = 0; j < 3; ++j) {
                const int n0 = (ng0 + j) * 16;
                v2f bv;
                bv.x = w1s[ka * 96 + n0 + lr];
                bv.y = w1s[(ka + 1) * 96 + n0 + lr];
                v8f& acc = (j == 0 ? acc0 : (j == 1 ? acc1 : acc2));
                acc = __builtin_amdgcn_wmma_f32_16x16x4_f32(
                    false, a, false, bv, (short)0, acc, false, false);
            }
        }
    }

    // bias + exact GELU (erf), store hidden to LDS
    #pragma unroll
    for (int j = 0; j < 3; ++j) {
        const v8f acc = (j == 0 ? acc0 : (j == 1 ? acc1 : acc2));
        const int col = (ng0 + j) * 16 + lr;
        const float bb = b1[col];
        #pragma unroll
        for (int r = 0; r < 8; ++r) {
            float v = acc[r] + bb;
            v = 0.5f * v * (1.0f + erff(v * 0.70710678118654752f));
            hid[(mt * 16 + 8 * half + r) * 98 + col] = v;
        }
    }
    __syncthreads();

    // ---------------- GEMM2: logits = hid[64,96] @ W2[96,96] ---------------
    v8f d0 = {}, d1 = {}, d2 = {};
    #pragma unroll 4
    for (int kk = 0; kk < 96; kk += 4) {
        const int ka = kk + 2 * half;
        v2f a = *(const v2f*)(hid + (mt * 16 + lr) * 98 + ka);
        #pragma unroll
        for (int j = 0; j < 3; ++j) {
            const int n0 = (ng0 + j) * 16;
            v2f bv;
            bv.x = w2s[ka * 96 + n0 + lr];
            bv.y = w2s[(ka + 1) * 96 + n0 + lr];
            v8f& acc = (j == 0 ? d0 : (j == 1 ? d1 : d2));
            acc = __builtin_amdgcn_wmma_f32_16x16x4_f32(
                false, a, false, bv, (short)0, acc, false, false);
        }
    }
    #pragma unroll
    for (int j = 0; j < 3; ++j) {
        const v8f acc = (j == 0 ? d0 : (j == 1 ? d1 : d2));
        const int col = (ng0 + j) * 16 + lr;
        const float bb = b2[col];
        #pragma unroll
        for (int r = 0; r < 8; ++r)
            w1s[(mt * 16 + 8 * half + r) * 96 + col] = acc[r] + bb;
    }
    __syncthreads();

    // softmax over 16 bases per (row, head), normalize in place in LDS
    for (int p = t; p < 64 * HH; p += 256) {
        const int row = p / HH, h = p % HH;
        float vals[KB];
        float mx = -3.402823466e38f;
        #pragma unroll
        for (int k = 0; k < KB; ++k) {
            vals[k] = w1s[row * 96 + k * HH + h];
            mx = fmaxf(mx, vals[k]);
        }
        float s = 0.0f;
        #pragma unroll
        for (int k = 0; k < KB; ++k) { vals[k] = __expf(vals[k] - mx); s += vals[k]; }
        const float inv = 1.0f / s;
        #pragma unroll
        for (int k = 0; k < KB; ++k) w1s[row * 96 + k * HH + h] = vals[k] * inv;
    }
    __syncthreads();
    // coalesced float4 writeback of the 64x96 prob block
    #pragma unroll
    for (int i = 0; i < 6; ++i) {
        int f = t + 256 * i;                // 1536 float4
        ((float4*)(mixo + (size_t)row0 * KH))[f] = ((const float4*)w1s)[f];
    }
}

// ---------------------------------------------------------------------------
// Kernel 2: per (b, m-tile of 32):
//   Phase A (VALU):  Wt[h][mt][n] = sum_k mix[b,n,k,h] * wb[k,n,m0+mt]  (150KB LDS)
//   Phase B (WMMA):  out[b,m,h*64+c] = sum_n Wt[h][m-m0][n] * x[b,n,h*64+c]
//   x[b] streamed through LDS in 28-row chunks (TDM); each B-operand load
//   feeds both mi WMMA tiles. 896 blocks x 256 threads.
// ---------------------------------------------------------------------------
__global__ __launch_bounds__(256) void mix_apply_kernel(
    const float* __restrict__ x, const float* __restrict__ wb,
    const float* __restrict__ mix, float* __restrict__ out)
{
    extern __shared__ float smem[];
    float* wt = smem;                       // [6][32][200] = 38400 floats
    float* xs = smem + 38400;               // [28][384] x chunk = 10752 floats

    const int t    = threadIdx.x;
    const int wave = t >> 5;
    const int lane = t & 31;
    const int b    = blockIdx.x / 7;
    const int m0   = (blockIdx.x % 7) * 32;
    const int mg   = m0 + lane;
    const bool mok = (mg < NTOK);

    // ---- Phase A: build dynamic weight tile in LDS ----
    const float* mixb = mix + (size_t)b * NTOK * KH;
    for (int n = wave; n < NTOK; n += 8) {
        float acc[HH] = {0.f, 0.f, 0.f, 0.f, 0.f, 0.f};
        const float* mrow = mixb + n * KH;
        if (n + 8 < NTOK)                   // warm next wb row (global_prefetch_b8)
            __builtin_prefetch(&wb[(size_t)(n + 8) * NTOK + mg], 0, 1);
        #pragma unroll 4
        for (int k = 0; k < KB; ++k) {
            const float wv = mok ? wb[((size_t)k * NTOK + n) * NTOK + mg] : 0.0f;
            #pragma unroll
            for (int h = 0; h < HH; ++h) acc[h] += mrow[k * HH + h] * wv;
        }
        #pragma unroll
        for (int h = 0; h < HH; ++h) wt[(h * 32 + lane) * 200 + n] = acc[h];
    }

    // ---- Phase B ----
    const int half = lane >> 4;
    const int lr   = lane & 15;
    const float* xb = x + (size_t)b * NTOK * CDIM;
    v8f acc[3][2] = {};                     // 3 (h,ni) combos x 2 mi tiles
    for (int ch = 0; ch < 7; ++ch) {
        const int n0 = ch * 28;
        __syncthreads();                    // wt ready (ch==0) / xs consumed (ch>0)
#ifdef HAVE_TDM
        if (wave == 0) {
            tdm_load_1d(K2_XS_OFF, xb + (size_t)n0 * CDIM, 28 * CDIM);
            __builtin_amdgcn_s_wait_tensorcnt(0);
        }
#else
        for (int i = 0; i < 42; ++i)
            xs[t + 256 * i] = xb[(size_t)n0 * CDIM + t + 256 * i];
#endif
        __syncthreads();
        #pragma unroll
        for (int j = 0; j < 3; ++j) {
            const int cmb = wave + 8 * j;   // 0..23 -> (h, ni)
            const int h  = cmb >> 2;
            const int ni = cmb & 3;
            const int c  = h * HD + ni * 16 + lr;
            const float* wth = wt + (h * 32) * 200 + n0;
            #pragma unroll
            for (int kk = 0; kk < 28; kk += 4) {
                const int ka = kk + 2 * half;
                v2f bv;                      // shared by both mi tiles
                bv.x = xs[ka * CDIM + c];
                bv.y = xs[(ka + 1) * CDIM + c];
                v2f a0 = *(const v2f*)(wth + lr * 200 + ka);
                v2f a1 = *(const v2f*)(wth + (16 + lr) * 200 + ka);
                acc[j][0] = __builtin_amdgcn_wmma_f32_16x16x4_f32(
                    false, a0, false, bv, (short)0, acc[j][0], false, false);
                acc[j][1] = __builtin_amdgcn_wmma_f32_16x16x4_f32(
                    false, a1, false, bv, (short)0, acc[j][1], false, false);
            }
        }
    }

    // ---- stores (uniform branch: 6 full tiles unmasked, last tile masked) --
    #pragma unroll
    for (int j = 0; j < 3; ++j) {
        const int cmb = wave + 8 * j;
        const int h  = cmb >> 2;
        const int ni = cmb & 3;
        const int c  = h * HD + ni * 16 + lr;
        if (m0 + 32 <= NTOK) {
            #pragma unroll
            for (int mi = 0; mi < 2; ++mi)
                #pragma unroll
                for (int rr = 0; rr < 8; ++rr) {
                    const int m = m0 + mi * 16 + 8 * half + rr;
                    out[((size_t)b * NTOK + m) * CDIM + c] = acc[j][mi][rr];
                }
        } else {
            #pragma unroll
            for (int mi = 0; mi < 2; ++mi)
                #pragma unroll
                for (int rr = 0; rr < 8; ++rr) {
                    const int m = m0 + mi * 16 + 8 * half + rr;
                    if (m < NTOK)
                        out[((size_t)b * NTOK + m) * CDIM + c] = acc[j][mi][rr];
                }
        }
    }
}

// ---------------------------------------------------------------------------
extern "C" void kernel_launch(void* const* d_in, const int* in_sizes, int n_in,
                              void* d_out, int out_size, void* d_ws, size_t ws_size,
                              hipStream_t stream)
{
    const float* x  = (const float*)d_in[0];  // [128,196,384]
    const float* W1 = (const float*)d_in[1];  // [384,96]
    const float* b1 = (const float*)d_in[2];  // [96]
    const float* W2 = (const float*)d_in[3];  // [96,96]
    const float* b2 = (const float*)d_in[4];  // [96]
    const float* wb = (const float*)d_in[5];  // [16,196,196]
    float* out   = (float*)d_out;             // [128,196,384]
    float* mixws = (float*)d_ws;              // [128*196, 96] softmaxed mix

    const size_t lds1 = (size_t)(64 * 68 + 64 * 96 + 64 * 98 + 96 * 96) * sizeof(float);
    mix_mlp_kernel<<<dim3(392), dim3(256), lds1, stream>>>(x, W1, b1, W2, b2, mixws);

    const size_t lds2 = (size_t)(38400 + 28 * CDIM) * sizeof(float);
    mix_apply_kernel<<<dim3(896), dim3(256), lds2, stream>>>(x, wb, mixws, out);
}